// SFE_25795573580099
// MI455X (gfx1250) — compile-verified
//
#include <hip/hip_runtime.h>
#include <hip/hip_bf16.h>
#include <math.h>

// ---------------------------------------------------------------------------
// Problem constants (match reference)
// ---------------------------------------------------------------------------
#define KNN      9
#define BATCH    16
#define PPTS     4096
#define NPTS     (BATCH * PPTS)      // 65536
#define CIN      9
#define COUT     32
#define CHUNK    1024                // candidate points staged in LDS per pass

typedef __attribute__((ext_vector_type(2))) float v2f;
typedef __attribute__((ext_vector_type(8))) float v8f;

// ---------------------------------------------------------------------------
// Kernel 1: brute-force kNN within each cloud.
// Rank key: d' = |q_j|^2 - 2 p_i . q_j   (drops constant |p_i|^2 -> same order)
// Candidates staged in LDS (float4 = xyz + |q|^2), broadcast reads.
// Top-9 kept in registers with compile-time-indexed branchless insert.
// ---------------------------------------------------------------------------
__global__ __launch_bounds__(256) void sfe_knn_kernel(
    const float* __restrict__ center, int* __restrict__ gidx)
{
    __shared__ float4 sp[CHUNK];
    const int tid = threadIdx.x;
    const int n = blockIdx.x * 256 + tid;
    const int cloudStart = (n >> 12) << 12;   // n/4096*4096, uniform per block

    const float qx = center[3 * n + 0];
    const float qy = center[3 * n + 1];
    const float qz = center[3 * n + 2];
    const float mx = -2.0f * qx, my = -2.0f * qy, mz = -2.0f * qz;

    float bd[KNN];
    int   bi[KNN];
#pragma unroll
    for (int k = 0; k < KNN; ++k) { bd[k] = 3.4e38f; bi[k] = cloudStart; }

    for (int c0 = 0; c0 < PPTS; c0 += CHUNK) {
        __syncthreads();
        for (int i = tid; i < CHUNK; i += 256) {
            const int g = cloudStart + c0 + i;
            const float x = center[3 * g + 0];
            const float y = center[3 * g + 1];
            const float z = center[3 * g + 2];
            sp[i] = make_float4(x, y, z, fmaf(x, x, fmaf(y, y, z * z)));
        }
        if (c0 + CHUNK < PPTS)   // pull next chunk toward caches while we scan
            __builtin_prefetch(&center[3 * (cloudStart + c0 + CHUNK) + (tid & 63) * 16], 0, 1);
        __syncthreads();

        for (int i = 0; i < CHUNK; ++i) {
            const float4 p = sp[i];
            float d = fmaf(mx, p.x, p.w);
            d = fmaf(my, p.y, d);
            d = fmaf(mz, p.z, d);
            if (d < bd[KNN - 1]) {           // rare: wave diverges only on inserts
                const int gi = cloudStart + c0 + i;
#pragma unroll
                for (int j = KNN - 1; j >= 1; --j) {
                    const bool lt = d < bd[j];
                    const bool fp = d < bd[j - 1];
                    const float nd = fp ? bd[j - 1] : d;
                    const int   ni = fp ? bi[j - 1] : gi;
                    bd[j] = lt ? nd : bd[j];
                    bi[j] = lt ? ni : bi[j];
                }
                if (d < bd[0]) { bd[0] = d; bi[0] = gi; }
            }
        }
    }
#pragma unroll
    for (int k = 0; k < KNN; ++k) gidx[n * KNN + k] = bi[k];
}

// ---------------------------------------------------------------------------
// Kernel 2: umbrella features per (point,k) + pass-1 of BN stats.
// One thread per point: gather 9 neighbors, azimuth sort (unrolled bubble
// network, all register-resident), triangle features, feat store, and
// x = feat@w1+b1 accumulated into sum/sumsq. Deterministic reduction:
// shuffle tree -> per-wave LDS -> per-block partial in ws.
// ---------------------------------------------------------------------------
__global__ __launch_bounds__(256) void sfe_feat_kernel(
    const float* __restrict__ center, const int* __restrict__ gidx,
    const float* __restrict__ w1, const float* __restrict__ b1,
    float* __restrict__ feat, float* __restrict__ bstats)
{
    __shared__ float sW1[CIN * COUT];
    __shared__ float sB1[COUT];
    __shared__ float sWave[8 * 64];
    const int tid = threadIdx.x;
    for (int i = tid; i < CIN * COUT; i += 256) sW1[i] = w1[i];
    if (tid < COUT) sB1[tid] = b1[tid];
    __syncthreads();

    const int n = blockIdx.x * 256 + tid;
    const float cx = center[3 * n + 0];
    const float cy = center[3 * n + 1];
    const float cz = center[3 * n + 2];

    float gx[KNN], gy[KNN], gz[KNN], ph[KNN];
#pragma unroll
    for (int k = 0; k < KNN; ++k) {
        const int g = gidx[n * KNN + k];
        const float x = center[3 * g + 0] - cx;
        const float y = center[3 * g + 1] - cy;
        const float z = center[3 * g + 2] - cz;
        gx[k] = x; gy[k] = y; gz[k] = z;
        // fixed rotate, columns 0 and 1 only (atan2 is order-equivalent to phi)
        const float r0 =  0.5f * x + 0.7071f * y - 0.5f * z;
        const float r1 = -0.5f * x + 0.7071f * y + 0.5f * z;
        ph[k] = atan2f(r1, r0);
    }

    // sort 9 triples by phi: fully unrolled bubble network (register-resident)
#pragma unroll
    for (int pass = 0; pass < KNN - 1; ++pass) {
#pragma unroll
        for (int j = 0; j < KNN - 1; ++j) {
            const float p0 = ph[j], p1 = ph[j + 1];
            const bool sw = p0 > p1;
            ph[j] = sw ? p1 : p0; ph[j + 1] = sw ? p0 : p1;
            float t;
            t = gx[j]; gx[j] = sw ? gx[j + 1] : t; gx[j + 1] = sw ? t : gx[j + 1];
            t = gy[j]; gy[j] = sw ? gy[j + 1] : t; gy[j + 1] = sw ? t : gy[j + 1];
            t = gz[j]; gz[j] = sw ? gz[j + 1] : t; gz[j + 1] = sw ? t : gz[j + 1];
        }
    }

    float xsum[COUT], xsq[COUT];
#pragma unroll
    for (int c = 0; c < COUT; ++c) { xsum[c] = 0.0f; xsq[c] = 0.0f; }

    float mask = 1.0f;
#pragma unroll
    for (int k = 0; k < KNN; ++k) {
        const int k2 = (k + 1) % KNN;            // constant after unroll
        const float ax = gx[k],  ay = gy[k],  az = gz[k];
        const float bx = gx[k2], by = gy[k2], bz = gz[k2];
        // triangle verts (0, s, s_roll): e1 = s, e2 = s_roll
        const float nx = ay * bz - az * by;
        const float ny = az * bx - ax * bz;
        const float nz = ax * by - ay * bx;
        const float nlen = sqrtf(nx * nx + ny * ny + nz * nz);
        const float inv = (nlen > 0.0f) ? (1.0f / nlen) : 1.0f;
        float ux = nx * inv, uy = ny * inv, uz = nz * inv;
        if (k == 0) mask = (ux > 0.0f) ? 1.0f : -1.0f;  // per-point first-term sign
        ux *= mask; uy *= mask; uz *= mask;
        const float ccx = (ax + bx) * (1.0f / 3.0f);
        const float ccy = (ay + by) * (1.0f / 3.0f);
        const float ccz = (az + bz) * (1.0f / 3.0f);
        float pos = (ux * ccx + uy * ccy + uz * ccz) * 0.5773502691896258f;
        const float dist = (sqrtf(ax * ax + ay * ay + az * az) +
                            sqrtf(bx * bx + by * by + bz * bz)) * (1.0f / 3.0f);
        const float area = 0.5f * nlen;
        // nan_to_num (NaN -> 0)
        ux  = (ux == ux)  ? ux  : 0.0f;
        uy  = (uy == uy)  ? uy  : 0.0f;
        uz  = (uz == uz)  ? uz  : 0.0f;
        pos = (pos == pos) ? pos : 0.0f;

        float f[CIN] = {ux, uy, uz, pos, ccx, ccy, ccz, dist, area};
        float* fp = feat + ((size_t)n * KNN + k) * CIN;
#pragma unroll
        for (int j = 0; j < CIN; ++j) fp[j] = f[j];

#pragma unroll
        for (int c = 0; c < COUT; ++c) {
            float x = sB1[c];
#pragma unroll
            for (int j = 0; j < CIN; ++j) x = fmaf(f[j], sW1[j * COUT + c], x);
            xsum[c] += x;
            xsq[c]  += x * x;
        }
    }

    // deterministic block reduction: shuffle tree (wave32) -> LDS -> partials
    const int lane = tid & 31, wv = tid >> 5;
#pragma unroll
    for (int c = 0; c < COUT; ++c) {
        float v = xsum[c];
#pragma unroll
        for (int off = 16; off >= 1; off >>= 1) v += __shfl_xor(v, off, 32);
        if (lane == 0) sWave[wv * 64 + c] = v;
        float q = xsq[c];
#pragma unroll
        for (int off = 16; off >= 1; off >>= 1) q += __shfl_xor(q, off, 32);
        if (lane == 0) sWave[wv * 64 + 32 + c] = q;
    }
    __syncthreads();
    if (tid < 64) {
        float a = 0.0f;
#pragma unroll
        for (int w = 0; w < 8; ++w) a += sWave[w * 64 + tid];
        bstats[blockIdx.x * 64 + tid] = a;
    }
}

// ---------------------------------------------------------------------------
// Kernel 3: fold block partials into BN scale/shift (fixed-order, deterministic)
// ---------------------------------------------------------------------------
__global__ void sfe_bn_finalize(const float* __restrict__ bstats,
                                const float* __restrict__ gamma,
                                const float* __restrict__ beta,
                                float* __restrict__ bnp)
{
    __shared__ float s[64];
    const int t = threadIdx.x;
    if (t < 64) {
        float a = 0.0f;
        for (int b = 0; b < 256; ++b) a += bstats[b * 64 + t];
        s[t] = a;
    }
    __syncthreads();
    if (t < 32) {
        const float invNK = 1.0f / ((float)NPTS * (float)KNN);
        const float mu  = s[t] * invNK;
        const float var = s[32 + t] * invNK - mu * mu;
        const float sc  = gamma[t] * rsqrtf(var + 1e-5f);
        bnp[t]      = sc;
        bnp[32 + t] = beta[t] - mu * sc;
    }
}

// ---------------------------------------------------------------------------
// Kernel 4: fused BN -> ReLU -> sum over K -> GEMM(32x32) via WMMA f32 16x16x4.
// Linearity trick: out[n] = (sum_k relu(bn(x[n,k]))) @ w2 + 9*b2, so the WMMA
// GEMM runs on N rows only. One wave = 16 points; lane pairs (l, l+16) split
// the 9 k-rows. Partials are staged through LDS (ds_store_b128 / ds_load_b64)
// so that fragment construction uses DS addressing instead of dynamically
// indexed register arrays (avoids v_cmp/v_cndmask waterfall lowering).
// Fragment layout per ISA: A 16x4 f32 (lanes<16: K=k0,k0+1; lanes>=16:
// K=k0+2,k0+3), B 4x16 (same K split, N=lane&15), C/D (lanes<16: M=v;
// lanes>=16: M=v+8).
// ---------------------------------------------------------------------------
__global__ __launch_bounds__(256) void sfe_mlp2_kernel(
    const float* __restrict__ feat,
    const float* __restrict__ w1, const float* __restrict__ b1,
    const float* __restrict__ bnp,
    const float* __restrict__ w2, const float* __restrict__ b2,
    float* __restrict__ out)
{
    __shared__ float sW1[CIN * COUT];
    __shared__ float sB1[COUT];
    __shared__ float sSc[COUT];
    __shared__ float sSh[COUT];
    __shared__ float sW2[COUT * COUT];
    __shared__ float sB2[COUT];
    __shared__ float sS[8][16][2][COUT];   // [wave][point][half][channel], 32 KB
    const int tid = threadIdx.x;
    for (int i = tid; i < CIN * COUT; i += 256) sW1[i] = w1[i];
    for (int i = tid; i < COUT * COUT; i += 256) sW2[i] = w2[i];
    if (tid < COUT) {
        sB1[tid] = b1[tid];
        sSc[tid] = bnp[tid];
        sSh[tid] = bnp[32 + tid];
        sB2[tid] = b2[tid];
    }
    __syncthreads();

    const int wv   = tid >> 5;
    const int lane = tid & 31;
    const int m    = lane & 15;
    const bool hi  = lane >= 16;
    const int half = hi ? 1 : 0;
    const int nbase = (blockIdx.x * 8 + wv) * 16;
    const int n     = nbase + m;

    // Phase 1: each half-lane accumulates BN/ReLU'd rows for its k-range.
    float part[COUT];
#pragma unroll
    for (int c = 0; c < COUT; ++c) part[c] = 0.0f;
    const int kbeg = hi ? 5 : 0;
    const int kend = hi ? 9 : 5;
    for (int k = kbeg; k < kend; ++k) {
        const float* fp = feat + ((size_t)n * KNN + k) * CIN;
        float f[CIN];
#pragma unroll
        for (int j = 0; j < CIN; ++j) f[j] = fp[j];
#pragma unroll
        for (int c = 0; c < COUT; ++c) {
            float x = sB1[c];
#pragma unroll
            for (int j = 0; j < CIN; ++j) x = fmaf(f[j], sW1[j * COUT + c], x);
            x = fmaf(x, sSc[c], sSh[c]);       // BN affine
            part[c] += fmaxf(x, 0.0f);         // ReLU + sum over K
        }
    }

    // Stage partials to LDS with b128 stores (constant-indexed register reads).
    float4* dst = (float4*)&sS[wv][m][half][0];
#pragma unroll
    for (int q = 0; q < COUT / 4; ++q)
        dst[q] = make_float4(part[4 * q + 0], part[4 * q + 1],
                             part[4 * q + 2], part[4 * q + 3]);
    __syncthreads();

    // Phase 2: D[16x16] += A[16x4] * B[4x16], 8 K-steps, 2 column tiles.
    // A-fragment channels come from LDS (pair loads), summing both halves.
    const float* aS0 = &sS[wv][m][0][0];
    const float* aS1 = &sS[wv][m][1][0];
    v8f acc0 = {0.f, 0.f, 0.f, 0.f, 0.f, 0.f, 0.f, 0.f};
    v8f acc1 = {0.f, 0.f, 0.f, 0.f, 0.f, 0.f, 0.f, 0.f};
    const int col = m;
    const int koff = hi ? 2 : 0;
#pragma unroll
    for (int ks = 0; ks < 8; ++ks) {
        const int c0 = 4 * ks + koff;          // even -> 8B-aligned LDS pair
        const float2 u = *(const float2*)(aS0 + c0);
        const float2 w = *(const float2*)(aS1 + c0);
        v2f a; a[0] = u.x + w.x; a[1] = u.y + w.y;
        v2f bA; bA[0] = sW2[c0 * COUT + col];      bA[1] = sW2[(c0 + 1) * COUT + col];
        v2f bB; bB[0] = sW2[c0 * COUT + 16 + col]; bB[1] = sW2[(c0 + 1) * COUT + 16 + col];
        acc0 = __builtin_amdgcn_wmma_f32_16x16x4_f32(false, a, false, bA,
                                                     (short)0, acc0, false, false);
        acc1 = __builtin_amdgcn_wmma_f32_16x16x4_f32(false, a, false, bB,
                                                     (short)0, acc1, false, false);
    }

    const int orow = nbase + (hi ? 8 : 0);
    const float bb0 = 9.0f * sB2[col];
    const float bb1 = 9.0f * sB2[16 + col];
#pragma unroll
    for (int v = 0; v < 8; ++v) {
        out[(orow + v) * COUT + col]      = acc0[v] + bb0;
        out[(orow + v) * COUT + 16 + col] = acc1[v] + bb1;
    }
}

// ---------------------------------------------------------------------------
// Host-side launch (graph-capture safe: kernels + stream only)
// ---------------------------------------------------------------------------
extern "C" void kernel_launch(void* const* d_in, const int* in_sizes, int n_in,
                              void* d_out, int out_size, void* d_ws, size_t ws_size,
                              hipStream_t stream) {
    const float* center = (const float*)d_in[0];
    const float* w1     = (const float*)d_in[1];
    const float* b1     = (const float*)d_in[2];
    const float* gamma  = (const float*)d_in[3];
    const float* beta   = (const float*)d_in[4];
    const float* w2     = (const float*)d_in[5];
    const float* b2     = (const float*)d_in[6];
    // d_in[7] = offset: uniform (B clouds of P points), not needed on device.
    float* out = (float*)d_out;

    // Workspace layout (~23.7 MB):
    //   [0)              gidx   : N*K ints
    //   [589824)         feat   : N*K*9 floats
    //   [+5308416)       bstats : 256 blocks * 64 floats
    //   [+16384)         bnp    : 64 floats (scale, shift)
    int*   gidx   = (int*)d_ws;
    float* fbase  = (float*)d_ws;
    float* feat   = fbase + (size_t)NPTS * KNN;                 // 589824
    float* bstats = feat + (size_t)NPTS * KNN * CIN;            // +5308416
    float* bnp    = bstats + 256 * 64;                          // +16384

    sfe_knn_kernel <<<NPTS / 256, 256, 0, stream>>>(center, gidx);
    sfe_feat_kernel<<<NPTS / 256, 256, 0, stream>>>(center, gidx, w1, b1, feat, bstats);
    sfe_bn_finalize<<<1, 64, 0, stream>>>(bstats, gamma, beta, bnp);
    sfe_mlp2_kernel<<<NPTS / 128, 256, 0, stream>>>(feat, w1, b1, bnp, w2, b2, out);
}